// ProtoLayer_10075993276528
// MI455X (gfx1250) — compile-verified
//
#include <hip/hip_runtime.h>

// Problem constants (from reference): t=32, WAY=64, SHOT=8, QUERY=16, c=1024
#define T_BATCH 32
#define WAY 64
#define SHOT 8
#define QUERY 16
#define C 1024
#define WQ (WAY * QUERY) /* 1024 query rows per batch */
#define WS (WAY * SHOT)  /* 512 support rows per batch */

#define KC 64          /* K-chunk staged in LDS */
#define MB 32          /* query rows per block */
#define LDS_STRIDE 68  /* 64 + 4 pad: rows stay 16B-aligned, banks conflict-free */

typedef __attribute__((ext_vector_type(2))) float v2f;
typedef __attribute__((ext_vector_type(8))) float v8f;

// ---------------------------------------------------------------------------
// CDNA5 async global->LDS copy (ASYNCcnt path). One B128 (16B) per lane:
//   VDST = per-lane LDS byte offset, VADDR = per-lane 64-bit global address.
// LDS offset = low 32 bits of the generic pointer (flat->LDS aperture rule).
// ---------------------------------------------------------------------------
__device__ __forceinline__ void async_copy_b128(void* lds_dst, const void* gsrc) {
    asm volatile("global_load_async_to_lds_b128 %0, %1, off"
                 :
                 : "v"((unsigned)(unsigned long long)lds_dst),
                   "v"((unsigned long long)gsrc)
                 : "memory");
}
__device__ __forceinline__ void wait_asynccnt0() {
    asm volatile("s_wait_asynccnt 0" ::: "memory");
}

// ---------------------------------------------------------------------------
// Kernel A: prototypes (mean over shots) + ||p||^2, one block per (t, way).
// ---------------------------------------------------------------------------
__global__ void __launch_bounds__(256)
proto_psq_kernel(const float* __restrict__ support,
                 float* __restrict__ proto,
                 float* __restrict__ psq) {
    __shared__ float red[256];
    const int tid = threadIdx.x;
    const int b   = blockIdx.x;       // t*WAY + w
    const int t   = b >> 6;
    const int w   = b & 63;
    const size_t sbase = ((size_t)t * WS + (size_t)w * SHOT) * C;
    const size_t pbase = ((size_t)t * WAY + (size_t)w) * C;

    float sq = 0.f;
    for (int i = tid; i < C; i += 256) {
        float s = 0.f;
#pragma unroll
        for (int sh = 0; sh < SHOT; ++sh)
            s += support[sbase + (size_t)sh * C + i];
        const float m = s * (1.0f / (float)SHOT);
        proto[pbase + i] = m;
        sq += m * m;
    }
    red[tid] = sq;
    __syncthreads();
    for (int off = 128; off > 0; off >>= 1) {
        if (tid < off) red[tid] += red[tid + off];
        __syncthreads();
    }
    if (tid == 0) psq[t * WAY + w] = red[0];
}

// ---------------------------------------------------------------------------
// Kernel B: dist = 2*Q.P^T - ||q||^2 - ||p||^2 via V_WMMA_F32_16X16X4_F32,
// double-buffered through LDS with async global->LDS B128 copies so the DMA
// of chunk k+1 overlaps the 16-WMMA stream on chunk k.
// Grid: (WQ/MB, T). Block: 256 threads = 8 waves; wave (mi,ni) owns the
// 16x16 tile at local rows mi*16..+15, ways ni*16..+15.
// ---------------------------------------------------------------------------
__global__ void __launch_bounds__(256)
proto_dist_wmma_kernel(const float* __restrict__ q,
                       const float* __restrict__ proto,
                       const float* __restrict__ psq,
                       float* __restrict__ out) {
    __shared__ float qsb[2][MB * LDS_STRIDE];   // 2 x  8.7 KB
    __shared__ float psb[2][WAY * LDS_STRIDE];  // 2 x 17.4 KB
    __shared__ float qsq[MB];

    const int tid   = threadIdx.x;
    const int lane  = tid & 31;
    const int wv    = tid >> 5;     // wave 0..7
    const int lhalf = lane >> 4;    // 0: lanes 0-15, 1: lanes 16-31
    const int l16   = lane & 15;
    const int t     = blockIdx.y;
    const int row0  = blockIdx.x * MB;

    const int mt = (wv & 1) * 16;   // local M tile base (0 or 16)
    const int nb = (wv >> 1) * 16;  // N (way) tile base (0,16,32,48)

    const float* qT = q     + (size_t)t * WQ  * C;
    const float* pT = proto + (size_t)t * WAY * C;

    // Per-thread tile coordinates for the cooperative async fills.
    const int qr0 = tid >> 4, qc = (tid & 15) * 4;     // Q: idx 0..255
    const int qr1 = qr0 + 16;                          // Q: idx 256..511

    v8f   acc  = {};
    float qacc = 0.f;               // per-row ||q||^2 partial (threads 0..31)

    // --- prologue: fill buffer 0 with chunk 0 ---
    {
        async_copy_b128(&qsb[0][qr0 * LDS_STRIDE + qc], &qT[(size_t)(row0 + qr0) * C + qc]);
        async_copy_b128(&qsb[0][qr1 * LDS_STRIDE + qc], &qT[(size_t)(row0 + qr1) * C + qc]);
#pragma unroll
        for (int u = 0; u < 4; ++u) {               // proto: 64x16 float4
            const int r = qr0 + u * 16;
            async_copy_b128(&psb[0][r * LDS_STRIDE + qc], &pT[(size_t)r * C + qc]);
        }
        wait_asynccnt0();
        __syncthreads();
    }

    int cur = 0;
    for (int k0 = 0; k0 < C; k0 += KC) {
        const int nxt = cur ^ 1;
        const int kn  = k0 + KC;

        // --- kick off DMA for the next chunk into the alternate buffer ---
        if (kn < C) {
            async_copy_b128(&qsb[nxt][qr0 * LDS_STRIDE + qc], &qT[(size_t)(row0 + qr0) * C + kn + qc]);
            async_copy_b128(&qsb[nxt][qr1 * LDS_STRIDE + qc], &qT[(size_t)(row0 + qr1) * C + kn + qc]);
#pragma unroll
            for (int u = 0; u < 4; ++u) {
                const int r = qr0 + u * 16;
                async_copy_b128(&psb[nxt][r * LDS_STRIDE + qc], &pT[(size_t)r * C + kn + qc]);
            }
        }

        // --- ||q||^2 partials straight out of LDS (HBM bytes read once) ---
        if (tid < MB) {
#pragma unroll 8
            for (int c = 0; c < KC; ++c) {
                const float v = qsb[cur][tid * LDS_STRIDE + c];
                qacc += v * v;
            }
        }

        // --- 16 x WMMA(K=4) on the current buffer ---
        // A frag (16x4 f32 layout): lanes 0-15 -> K0,K1 ; lanes 16-31 -> K2,K3
        const float* aBase = &qsb[cur][(mt + l16) * LDS_STRIDE + 2 * lhalf];
        const float* bBase = &psb[cur][(nb + l16) * LDS_STRIDE + 2 * lhalf];
#pragma unroll
        for (int kk = 0; kk < KC; kk += 4) {
            const v2f a = *(const v2f*)(aBase + kk);
            const v2f b = *(const v2f*)(bBase + kk);
            acc = __builtin_amdgcn_wmma_f32_16x16x4_f32(
                /*neg_a=*/false, a, /*neg_b=*/false, b,
                /*c_mod=*/(short)0, acc, /*reuse_a=*/false, /*reuse_b=*/false);
        }

        wait_asynccnt0();   // our async writes to nxt have landed in LDS
        __syncthreads();    // all waves done reading cur / writing nxt
        cur = nxt;
    }

    if (tid < MB) qsq[tid] = qacc;
    __syncthreads();

    // Epilogue: C/D layout -> VGPR i holds (M = i + 8*lhalf, N = l16)
    const float pv = psq[t * WAY + nb + l16];
#pragma unroll
    for (int i = 0; i < 8; ++i) {
        const int mloc = mt + i + lhalf * 8;
        const float d  = 2.0f * acc[i] - qsq[mloc] - pv;
        out[((size_t)t * WQ + row0 + mloc) * WAY + nb + l16] = d;
    }
}

// ---------------------------------------------------------------------------
extern "C" void kernel_launch(void* const* d_in, const int* in_sizes, int n_in,
                              void* d_out, int out_size, void* d_ws, size_t ws_size,
                              hipStream_t stream) {
    const float* query   = (const float*)d_in[0]; // [32,1024,1024]
    const float* support = (const float*)d_in[1]; // [32, 512,1024]
    float* out = (float*)d_out;                   // [32,1024,  64]

    // workspace: proto [32*64*1024] f32 (8 MB) + psq [32*64] f32 (8 KB)
    float* proto = (float*)d_ws;
    float* psq   = proto + (size_t)T_BATCH * WAY * C;

    proto_psq_kernel<<<dim3(T_BATCH * WAY), dim3(256), 0, stream>>>(
        support, proto, psq);

    dim3 grid(WQ / MB, T_BATCH); // (32, 32)
    proto_dist_wmma_kernel<<<grid, dim3(256), 0, stream>>>(
        query, proto, psq, out);
}